// DifferentiableDrone_4380866642557
// MI455X (gfx1250) — compile-verified
//
#include <hip/hip_runtime.h>

// Drone RK4 dynamics, B=2,000,000 independent rows.
// Memory-streaming kernel: one wave32 lane per row, B128 non-temporal
// loads/stores, hardware transcendentals (v_sin/v_cos/v_rcp/v_sqrt).

#define DT_   0.01f
#define KD_   0.05f

typedef float v4f __attribute__((ext_vector_type(4)));

__device__ __forceinline__ void drone_dyn(
    const float* __restrict__ s,              // 12-float stage state
    float u0, float t0, float t1, float t2,   // thrust + torques
    float I0, float I1, float I2,
    float invI0, float invI1, float invI2,
    float gx, float gy, float gz,
    float* __restrict__ k)                    // 12-float derivative out
{
    const float vx = s[3], vy = s[4], vz = s[5];
    const float roll = s[6], pitch = s[7], yaw = s[8];
    const float w0 = s[9], w1 = s[10], w2 = s[11];

    float s_r, c_r, s_p, c_p, s_y, c_y;
    __sincosf(roll,  &s_r, &c_r);   // v_sin_f32 / v_cos_f32
    __sincosf(pitch, &s_p, &c_p);
    __sincosf(yaw,   &s_y, &c_y);

    // Only column 2 of the rotation matrix is ever used.
    const float R02 = c_y * s_p * c_r + s_y * s_r;
    const float R12 = s_y * s_p * c_r - c_y * s_r;
    const float R22 = c_p * c_r;

    // Drag: -K * |v| * v  (norm clamped at 1e-8)
    float vn = __builtin_sqrtf(vx * vx + vy * vy + vz * vz);
    vn = fmaxf(vn, 1e-8f);
    const float df = -KD_ * vn;

    // d(pos)/dt = vel
    k[0] = vx; k[1] = vy; k[2] = vz;
    // d(vel)/dt = (thrust_world + drag + g) / M, M == 1
    k[3] = fmaf(R02, u0, fmaf(df, vx, gx));
    k[4] = fmaf(R12, u0, fmaf(df, vy, gy));
    k[5] = fmaf(R22, u0, fmaf(df, vz, gz));

    // omega_dot = (tau - omega x (I*omega)) / I
    const float wI0 = w0 * I0, wI1 = w1 * I1, wI2 = w2 * I2;
    const float cx = w1 * wI2 - w2 * wI1;
    const float cy = w2 * wI0 - w0 * wI2;
    const float cz = w0 * wI1 - w1 * wI0;
    k[9]  = (t0 - cx) * invI0;
    k[10] = (t1 - cy) * invI1;
    k[11] = (t2 - cz) * invI2;

    // Euler-angle kinematics; single v_rcp_f32 shared by 3 reference divisions.
    const float cp_safe = (__builtin_fabsf(c_p) < 1e-6f) ? 1e-6f : c_p;
    const float inv_cp  = __builtin_amdgcn_rcpf(cp_safe);
    const float t_p     = s_p * inv_cp;
    k[6] = fmaf(s_r * t_p, w1, fmaf(c_r * t_p, w2, w0));
    k[7] = c_r * w1 - s_r * w2;
    k[8] = fmaf(s_r * inv_cp, w1, (c_r * inv_cp) * w2);
}

__global__ __launch_bounds__(256) void drone_rk4_kernel(
    const float* __restrict__ state,
    const float* __restrict__ u,
    const float* __restrict__ Iv,
    const float* __restrict__ gv,
    float* __restrict__ out,
    int B)
{
    const int i = blockIdx.x * blockDim.x + threadIdx.x;
    if (i >= B) return;

    // Uniform constants -> scalar loads.
    const float I0 = Iv[0], I1 = Iv[1], I2 = Iv[2];
    // v_rcp_f32 (~1 ulp) instead of full IEEE div sequence: uniform, benign
    // values (0.01/0.01/0.02); drops ~30 VALU ops of div_scale/div_fmas chain.
    const float invI0 = __builtin_amdgcn_rcpf(I0);
    const float invI1 = __builtin_amdgcn_rcpf(I1);
    const float invI2 = __builtin_amdgcn_rcpf(I2);
    const float gx = gv[0], gy = gv[1], gz = gv[2];

    // Row loads: 3 x b128 (state) + 1 x b128 (u), all 16B aligned & coalesced.
    const v4f* sp = reinterpret_cast<const v4f*>(state) + (size_t)i * 3;
    const v4f a = __builtin_nontemporal_load(sp + 0);
    const v4f b = __builtin_nontemporal_load(sp + 1);
    const v4f c = __builtin_nontemporal_load(sp + 2);
    const v4f uu = __builtin_nontemporal_load(reinterpret_cast<const v4f*>(u) + i);

    float s[12] = { a.x, a.y, a.z, a.w,  b.x, b.y, b.z, b.w,  c.x, c.y, c.z, c.w };
    const float u0 = uu.x, t0 = uu.y, t1 = uu.z, t2 = uu.w;

    float k1[12], k2[12], k3[12], k4[12], tmp[12];

    drone_dyn(s, u0, t0, t1, t2, I0, I1, I2, invI0, invI1, invI2, gx, gy, gz, k1);
    #pragma unroll
    for (int j = 0; j < 12; ++j) tmp[j] = fmaf(0.5f * DT_, k1[j], s[j]);

    drone_dyn(tmp, u0, t0, t1, t2, I0, I1, I2, invI0, invI1, invI2, gx, gy, gz, k2);
    #pragma unroll
    for (int j = 0; j < 12; ++j) tmp[j] = fmaf(0.5f * DT_, k2[j], s[j]);

    drone_dyn(tmp, u0, t0, t1, t2, I0, I1, I2, invI0, invI1, invI2, gx, gy, gz, k3);
    #pragma unroll
    for (int j = 0; j < 12; ++j) tmp[j] = fmaf(DT_, k3[j], s[j]);

    drone_dyn(tmp, u0, t0, t1, t2, I0, I1, I2, invI0, invI1, invI2, gx, gy, gz, k4);

    float o[12];
    #pragma unroll
    for (int j = 0; j < 12; ++j) {
        const float ksum = k1[j] + 2.0f * (k2[j] + k3[j]) + k4[j];
        o[j] = fmaf(DT_ / 6.0f, ksum, s[j]);
    }

    // 3 x b128 non-temporal stores (one-shot stream; keep L2 clean).
    v4f* op = reinterpret_cast<v4f*>(out) + (size_t)i * 3;
    v4f o0 = { o[0], o[1], o[2],  o[3] };
    v4f o1 = { o[4], o[5], o[6],  o[7] };
    v4f o2 = { o[8], o[9], o[10], o[11] };
    __builtin_nontemporal_store(o0, op + 0);
    __builtin_nontemporal_store(o1, op + 1);
    __builtin_nontemporal_store(o2, op + 2);
}

extern "C" void kernel_launch(void* const* d_in, const int* in_sizes, int n_in,
                              void* d_out, int out_size, void* d_ws, size_t ws_size,
                              hipStream_t stream) {
    const float* state = (const float*)d_in[0];  // (B, 12) f32
    const float* u     = (const float*)d_in[1];  // (B, 4)  f32
    const float* Iv    = (const float*)d_in[2];  // (3,)    f32
    const float* gv    = (const float*)d_in[3];  // (3,)    f32
    float* out = (float*)d_out;                  // (B, 12) f32

    const int B = in_sizes[0] / 12;
    const int threads = 256;
    const int blocks = (B + threads - 1) / threads;
    drone_rk4_kernel<<<blocks, threads, 0, stream>>>(state, u, Iv, gv, out, B);
}